// LiftSplatShoot_18519898980434
// MI455X (gfx1250) — compile-verified
//
#include <hip/hip_runtime.h>

// ---------------------------------------------------------------------------
// Lift-Splat-Shoot BEV splat for MI455X (gfx1250).
// Scatter/atomic-bound (~84M f32 atomics, BEV output resident in 192MB L2).
// Per (cam, image-row) block: async-DMA the 103x50 head slice into LDS
// (features transposed to [w][f] on the fly), softmax depth logits in LDS,
// project points with v_wmma_f32_16x16x4_f32, broadcast per-point cell/weight
// via v_readlane (scalar), and scatter 64 features/point with f32 atomics.
// ---------------------------------------------------------------------------

#define B_    4
#define N_    6
#define HF    28
#define WF    50
#define DB    39
#define FEAT_ 64
#define CH    (DB + FEAT_)        // 103 channels
#define HB    188
#define WB    126
#define HBWB  (HB * WB)           // 23688
#define NPTS  (WF * DB)           // 1950 points per (cam,row)
#define NGRP  ((NPTS + 15) / 16)  // 122 groups of 16 points

typedef __attribute__((ext_vector_type(2))) float v2f;
typedef __attribute__((ext_vector_type(8))) float v8f;
typedef __attribute__((address_space(3))) float lds_f;

__device__ __forceinline__ unsigned lds_off(const float* p) {
  return (unsigned)(unsigned long long)(lds_f*)p;
}
__device__ __forceinline__ float readlane_f(float v, int l) {
  return __int_as_float(__builtin_amdgcn_readlane(__float_as_int(v), l));
}

__global__ __launch_bounds__(256) void zero_kernel(float4* __restrict__ out, int n4) {
  int i = blockIdx.x * 256 + threadIdx.x;
  if (i < n4) out[i] = make_float4(0.f, 0.f, 0.f, 0.f);
}

__global__ __launch_bounds__(256) void splat_kernel(
    const float* __restrict__ head,   // (24, 103, 28, 50)
    const float* __restrict__ Kmat,   // (4, 6, 3, 3)
    const float* __restrict__ Tmat,   // (4, 6, 4, 4) car2cam
    float* __restrict__ out) {        // (4, 64, 188, 126), pre-zeroed
  __shared__ float sdep[DB * WF];     // depth logits -> softmax weights [d][w]
  __shared__ float sfeat[WF * FEAT_]; // features transposed to [w][f]
  __shared__ float sT[8];             // rows 0,1 of T = inv(car2cam) @ inv_K4

  const int bn = blockIdx.x / HF;
  const int h  = blockIdx.x % HF;
  const int b  = bn / N_;

  // -------- async copy head[bn, :, h, :] into LDS (features transposed) ----
  const float* gbase   = head + (size_t)bn * (CH * HF * WF) + (size_t)h * WF;
  const unsigned depB  = lds_off(sdep);
  const unsigned featB = lds_off(sfeat);
  for (int l = threadIdx.x; l < CH * WF; l += 256) {
    int c = l / WF;
    int w = l - c * WF;
    unsigned goff  = (unsigned)((c * (HF * WF) + w) * 4);
    unsigned laddr = (c < DB) ? (depB  + (unsigned)((c * WF + w) * 4))
                              : (featB + (unsigned)((w * FEAT_ + (c - DB)) * 4));
    asm volatile("global_load_async_to_lds_b32 %0, %1, %2"
                 :: "v"(laddr), "v"(goff), "s"(gbase) : "memory");
  }

  // Overlap with the DMA: thread 0 builds rows 0,1 of
  // T = inv(car2cam) @ blkdiag(inv(K), 1).  Rigid inverse + 3x3 adjugate.
  if (threadIdx.x == 0) {
    const float* k = Kmat + bn * 9;
    const float* t = Tmat + bn * 16;
    float a = k[0], bb = k[1], c = k[2], d = k[3], e = k[4], f = k[5],
          g = k[6], hh = k[7], ii = k[8];
    float A0 = e * ii - f * hh, B0 = -(d * ii - f * g), C0 = d * hh - e * g;
    float idet = 1.f / (a * A0 + bb * B0 + c * C0);
    float ik[9] = { A0 * idet, -(bb * ii - c * hh) * idet, (bb * f - c * e) * idet,
                    B0 * idet,  (a * ii - c * g)  * idet, -(a * f - c * d) * idet,
                    C0 * idet, -(a * hh - bb * g) * idet, (a * e - bb * d) * idet };
    for (int r = 0; r < 2; ++r) {
      float c0 = t[0 * 4 + r], c1 = t[1 * 4 + r], c2 = t[2 * 4 + r];
      float c3 = -(c0 * t[3] + c1 * t[7] + c2 * t[11]);
      sT[r * 4 + 0] = c0 * ik[0] + c1 * ik[3] + c2 * ik[6];
      sT[r * 4 + 1] = c0 * ik[1] + c1 * ik[4] + c2 * ik[7];
      sT[r * 4 + 2] = c0 * ik[2] + c1 * ik[5] + c2 * ik[8];
      sT[r * 4 + 3] = c3;
    }
  }

  asm volatile("s_wait_asynccnt 0x0" ::: "memory");
  __syncthreads();

  // -------- per-pixel softmax over DB depth logits, in place in LDS --------
  if (threadIdx.x < WF) {
    int w = threadIdx.x;
    float mx = sdep[w];
    for (int c = 1; c < DB; ++c) mx = fmaxf(mx, sdep[c * WF + w]);
    float s = 0.f;
    for (int c = 0; c < DB; ++c) {
      float e = __expf(sdep[c * WF + w] - mx);
      sdep[c * WF + w] = e;
      s += e;
    }
    float inv = __builtin_amdgcn_rcpf(s);
    for (int c = 0; c < DB; ++c) sdep[c * WF + w] *= inv;
  }
  __syncthreads();

  // -------- geometry via WMMA f32 16x16x4 + feature scatter ----------------
  const int  lane = threadIdx.x & 31;
  const int  wv   = threadIdx.x >> 5;   // wave id 0..7
  const int  n    = lane & 15;
  const bool hiH  = lane >= 16;

  // A-matrix (16x4 f32): row 0 = T row0, row 1 = T row1, rows 2..15 = 0.
  // Layout: lane m (lo half) holds K=0,1; lane m+16 holds K=2,3.
  v2f av;
  av.x = (n == 0) ? (hiH ? sT[2] : sT[0]) : ((n == 1) ? (hiH ? sT[6] : sT[4]) : 0.f);
  av.y = (n == 0) ? (hiH ? sT[3] : sT[1]) : ((n == 1) ? (hiH ? sT[7] : sT[5]) : 0.f);

  const float vc = (float)h * 16.f + 8.f;      // image v coord for this row
  const size_t bBase = (size_t)b * FEAT_ * HBWB;
  const int voff0 = (2 * lane) * HBWB;         // feature channel offsets
  const int voff1 = voff0 + HBWB;

  for (int g = wv; g < NGRP; g += 8) {  // wave-uniform trip count per wave
    int  p  = g * 16 + n;               // point index = w*DB + d
    bool pv = p < NPTS;
    int  pc = pv ? p : 0;
    int  w  = pc / DB;
    int  d  = pc - w * DB;
    float dep = 3.f + 2.f * (float)d;   // linspace(3, 79, 39)
    float u   = (float)w * 16.f + 8.f;

    // B-matrix (4x16 f32): column n = (u*d, v*d, d, 1).
    v2f bv;
    bv.x = hiH ? dep : u * dep;
    bv.y = hiH ? 1.f : vc * dep;

    v8f acc = {0.f, 0.f, 0.f, 0.f, 0.f, 0.f, 0.f, 0.f};
    acc = __builtin_amdgcn_wmma_f32_16x16x4_f32(
        false, av, false, bv, (short)0, acc, false, false);

    // D layout: VGPR0 lanes 0-15 = px, VGPR1 = py.
    float px = acc[0];
    float py = acc[1];
    int bx = (int)(px * 1.25f);                 // 1/0.8 == 1.25 exactly
    int by = (int)((py + 50.4f) * 1.25f);
    bool ok = pv && !hiH && (bx >= 0) && (bx < HB) && (by >= 0) && (by < WB);
    float wt  = ok ? sdep[d * WF + w] : 0.f;    // unconditional-safe index
    int  cell = bx * WB + by;

    // Broadcast each point's (cell, weight, pixel) to SGPRs via readlane;
    // all 32 lanes then scatter 2 features each (one ds_load_b64 + 2 atomics).
#pragma unroll
    for (int kk = 0; kk < 16; ++kk) {
      float wk = readlane_f(wt, kk);
      if (wk != 0.f) {                          // scalar branch
        int ck = __builtin_amdgcn_readlane(cell, kk);
        int pw = __builtin_amdgcn_readlane(w, kk);
        const float2 f01 = *(const float2*)&sfeat[pw * FEAT_ + 2 * lane];
        float* p0 = out + bBase + (size_t)ck;
        atomicAdd(p0 + voff0, wk * f01.x);
        atomicAdd(p0 + voff1, wk * f01.y);
      }
    }
  }
}

extern "C" void kernel_launch(void* const* d_in, const int* in_sizes, int n_in,
                              void* d_out, int out_size, void* d_ws, size_t ws_size,
                              hipStream_t stream) {
  const float* head = (const float*)d_in[0];   // (24,103,28,50)
  const float* Kmat = (const float*)d_in[1];   // (4,6,3,3)
  const float* Tmat = (const float*)d_in[2];   // (4,6,4,4)
  // d_in[3] (frustum) is recomputed analytically on device.
  float* out = (float*)d_out;                  // (4,64,188,126)

  int n4 = out_size / 4;
  zero_kernel<<<(n4 + 255) / 256, 256, 0, stream>>>((float4*)out, n4);
  splat_kernel<<<B_ * N_ * HF, 256, 0, stream>>>(head, Kmat, Tmat, out);
}